// CropProposals_13829794693462
// MI455X (gfx1250) — compile-verified
//
#include <hip/hip_runtime.h>
#include <stdint.h>

// Problem constants (match reference)
#define BB 4
#define CCH 128
#define DD 24
#define HH 24
#define WW 24
#define PP 64

#define CPW 8          // channels per wave
#define DCHUNK 5       // D-slices per TDM tile
#define MAXCROP 23     // max crop length per axis
#define CHUNK_ELEMS (DCHUNK * MAXCROP * MAXCROP)   // 2645 floats = 10.58 KB

typedef uint32_t u32x4 __attribute__((ext_vector_type(4)));
typedef uint32_t u32x8 __attribute__((ext_vector_type(8)));

#define NEG_INF (-3.402823466e+38f)

// Issue one TDM load: a [dcnt x nh x nw] fp32 tile (row stride WW, plane
// stride HH*WW) starting at global byte address gaddr, packed densely into
// LDS at byte offset lds_off. Tracked by TENSORcnt.
__device__ __forceinline__ void tdm_load_tile(uint32_t lds_off, uint64_t gaddr,
                                              int nw, int nh, int dcnt) {
  // ---- D# group 0 (4 dwords) ----
  u32x4 g0;
  g0.x = 1u;                                    // count=1, user descriptor
  g0.y = lds_off;                               // lds_addr (bytes)
  g0.z = (uint32_t)gaddr;                       // global_addr[31:0]
  g0.w = (uint32_t)((gaddr >> 32) & 0x1FFFFFFu) // global_addr[56:32]
         | (2u << 30);                          // type = 2 ("image")
  // ---- D# group 1 (8 dwords) ----
  const uint32_t td0 = 0x00FFFFFFu;             // huge tensor dims: crop is
  const uint32_t td1 = 0x00FFFFFFu;             // always in-bounds, never OOB
  const uint32_t st0 = (uint32_t)WW;            // dim0 stride = 24 elems
  const uint32_t st1 = (uint32_t)(HH * WW);     // dim1 stride = 576 elems
  u32x8 g1;
  g1.s0 = (2u << 16);                           // wg_mask=0, data_size=4B
  g1.s1 = (td0 & 0xFFFFu) << 16;                // tensor_dim0[15:0]
  g1.s2 = ((td0 >> 16) & 0xFFFFu) | ((td1 & 0xFFFFu) << 16);
  g1.s3 = ((td1 >> 16) & 0xFFFFu) | ((uint32_t)nw << 16);   // tile_dim0
  g1.s4 = (uint32_t)nh | ((uint32_t)dcnt << 16);            // tile_dim1,2
  g1.s5 = st0;                                  // tensor_dim0_stride[31:0]
  g1.s6 = 0u | ((st1 & 0xFFFFu) << 16);         // st0[47:32]=0, st1[15:0]
  g1.s7 = (st1 >> 16);                          // st1[47:16]
  // ---- D# group 2 (4 dwords), iterate_enable=0, tile_dim3=0 ----
  u32x4 g2;
  g2.x = 0x00FFFFFFu;                           // tensor_dim2 (huge)
  g2.y = 0u;                                    // tensor_dim3 (unused)
  g2.z = (uint32_t)(DD * HH * WW);              // tensor_dim2_stride[31:0]
  g2.w = 0u;                                    // stride hi=0, tile_dim3=0
  // ---- D# group 3 (4 dwords): zeros (dims 3/4 unused) ----
  u32x4 g3 = {0u, 0u, 0u, 0u};

  asm volatile("tensor_load_to_lds %0, %1, %2, %3"
               :
               : "s"(g0), "s"(g1), "s"(g2), "s"(g3)
               : "memory");
}

__global__ __launch_bounds__(32) void
crop_pool_tdm_kernel(const float* __restrict__ fm,
                     const int* __restrict__ corners,
                     const int* __restrict__ scale_p,
                     float* __restrict__ out) {
  __shared__ float lbuf[2][CHUNK_ELEMS];

  const int lane = threadIdx.x;        // wave32: lane id
  const int bp   = blockIdx.x;         // proposal index: b*PP + p
  const int cgrp = blockIdx.y;         // channel group (16 groups of CPW)

  const int b = bp / PP;

  const int scale = __builtin_amdgcn_readfirstlane(scale_p[0]);

  // Crop bounds per axis (axis 0->D, 1->H, 2->W), reference semantics:
  // p1 = clip(c0/scale, 0, 21); p2 = (c1/scale - p1 >= 2) ? c1/scale : p1+2
  int s[3], e[3];
#pragma unroll
  for (int a = 0; a < 3; ++a) {
    int c0 = __builtin_amdgcn_readfirstlane(corners[(bp * 2 + 0) * 3 + a]);
    int c1 = __builtin_amdgcn_readfirstlane(corners[(bp * 2 + 1) * 3 + a]);
    int p1 = c0 / scale;
    p1 = p1 < 0 ? 0 : (p1 > 21 ? 21 : p1);
    int p2 = c1 / scale;
    if (p2 - p1 < 2) p2 = p1 + 2;
    s[a] = p1;
    e[a] = p2;
  }
  const int sx = s[0], ex = e[0], nx = ex - sx;   // D axis
  const int sy = s[1], ny = e[1] - sy;            // H axis
  const int sz = s[2], nw = e[2] - sz;            // W axis

  // Adaptive-pool region bounds: region0 = [s, s+ceil(n/2)),
  //                              region1 = [s+floor(n/2), e)
  const int x0e = sx + ((nx + 1) >> 1);
  const int x1s = sx + (nx >> 1);
  const int y0e = sy + ((ny + 1) >> 1);
  const int y1s = sy + (ny >> 1);
  const int z0n = (nw + 1) >> 1;   // lane-local: region0 lanes [0, z0n)
  const int z1s = nw >> 1;         // lane-local: region1 lanes [z1s, nw)

  const bool wvalid = lane < nw;
  const bool in_z0  = lane < z0n;
  const bool in_z1  = (lane >= z1s) && wvalid;

  // LDS byte offsets of the two tile buffers (generic-ptr low 32 bits == LDS offset)
  const uint32_t lds_off0 = (uint32_t)(uintptr_t)(&lbuf[0][0]);
  const uint32_t lds_off1 = (uint32_t)(uintptr_t)(&lbuf[1][0]);

  const int nchunks = (nx + DCHUNK - 1) / DCHUNK;

  for (int k = 0; k < CPW; ++k) {
    const int c = cgrp * CPW + k;
    const uint64_t chan_base =
        (uint64_t)(uintptr_t)fm +
        4ull * ((((uint64_t)b * CCH + (uint64_t)c) * DD + (uint64_t)sx) *
                    (uint64_t)(HH * WW) +
                (uint64_t)(sy * WW + sz));

    // Prime the pipeline: chunk 0 -> buffer 0.
    {
      const int dcnt0 = nx < DCHUNK ? nx : DCHUNK;
      tdm_load_tile(lds_off0, chan_base, nw, ny, dcnt0);
    }

    float acc[8];
#pragma unroll
    for (int i = 0; i < 8; ++i) acc[i] = NEG_INF;

    for (int ck = 0; ck < nchunks; ++ck) {
      const int dbeg = ck * DCHUNK;
      const int dcnt = (nx - dbeg) < DCHUNK ? (nx - dbeg) : DCHUNK;

      if (ck + 1 < nchunks) {
        // Issue next chunk into the other buffer, then wait for current.
        const int nbeg = dbeg + DCHUNK;
        const int ncnt = (nx - nbeg) < DCHUNK ? (nx - nbeg) : DCHUNK;
        tdm_load_tile((ck & 1) ? lds_off0 : lds_off1,
                      chan_base + 4ull * (uint64_t)nbeg * (uint64_t)(HH * WW),
                      nw, ny, ncnt);
        __builtin_amdgcn_s_wait_tensorcnt(1);   // current chunk landed
      } else {
        __builtin_amdgcn_s_wait_tensorcnt(0);   // drain
      }

      const float* bufk = lbuf[ck & 1];
      for (int dl = 0; dl < dcnt; ++dl) {
        const int d = sx + dbeg + dl;
        const bool xm0 = d < x0e;     // d >= sx always holds
        const bool xm1 = d >= x1s;
        int rowbase = dl * ny * nw;
        for (int hl = 0; hl < ny; ++hl) {
          const int h = sy + hl;
          const bool ym0 = h < y0e;
          const bool ym1 = h >= y1s;
          const float v  = wvalid ? bufk[rowbase + lane] : NEG_INF;
          const float v0 = in_z0 ? v : NEG_INF;
          const float v1 = in_z1 ? v : NEG_INF;
          if (xm0 && ym0) { acc[0] = fmaxf(acc[0], v0); acc[1] = fmaxf(acc[1], v1); }
          if (xm0 && ym1) { acc[2] = fmaxf(acc[2], v0); acc[3] = fmaxf(acc[3], v1); }
          if (xm1 && ym0) { acc[4] = fmaxf(acc[4], v0); acc[5] = fmaxf(acc[5], v1); }
          if (xm1 && ym1) { acc[6] = fmaxf(acc[6], v0); acc[7] = fmaxf(acc[7], v1); }
          rowbase += nw;
        }
      }
    }

    // Wave32 cross-lane max reduction for each of the 8 octants.
#pragma unroll
    for (int i = 0; i < 8; ++i) {
      float a = acc[i];
      a = fmaxf(a, __shfl_xor(a, 16, 32));
      a = fmaxf(a, __shfl_xor(a, 8, 32));
      a = fmaxf(a, __shfl_xor(a, 4, 32));
      a = fmaxf(a, __shfl_xor(a, 2, 32));
      a = fmaxf(a, __shfl_xor(a, 1, 32));
      acc[i] = a;
    }

    if (lane == 0) {
      float* o = out + ((uint64_t)bp * CCH + (uint64_t)c) * 8ull;
#pragma unroll
      for (int i = 0; i < 8; ++i) o[i] = acc[i];   // [ox][oy][oz] packed
    }
  }
}

extern "C" void kernel_launch(void* const* d_in, const int* in_sizes, int n_in,
                              void* d_out, int out_size, void* d_ws,
                              size_t ws_size, hipStream_t stream) {
  (void)in_sizes; (void)n_in; (void)out_size; (void)d_ws; (void)ws_size;
  const float* fm      = (const float*)d_in[0];
  const int*   corners = (const int*)d_in[1];
  const int*   scale   = (const int*)d_in[2];
  float*       out     = (float*)d_out;

  dim3 grid(BB * PP, CCH / CPW);   // (256, 16) single-wave workgroups
  crop_pool_tdm_kernel<<<grid, 32, 0, stream>>>(fm, corners, scale, out);
}